// InvoiceGCN_75703093559494
// MI455X (gfx1250) — compile-verified
//
#include <hip/hip_runtime.h>
#include <hip/hip_bf16.h>

typedef __attribute__((ext_vector_type(2))) float v2f;
typedef __attribute__((ext_vector_type(8))) float v8f;

// ---------------------------------------------------------------------------
// Degree / normalization kernels
// ---------------------------------------------------------------------------
__global__ void gcn_init_deg(float* __restrict__ deg, int n) {
  int i = blockIdx.x * blockDim.x + threadIdx.x;
  if (i < n) deg[i] = 2.0f;  // SELF_LOOP_W
}

__global__ void gcn_accum_deg(const long long* __restrict__ dst,
                              const float* __restrict__ ew,
                              float* __restrict__ deg, int e) {
  int i = blockIdx.x * blockDim.x + threadIdx.x;
  if (i < e) atomicAdd(&deg[(long)dst[i]], ew[i]);
}

__global__ void gcn_rsqrt(float* __restrict__ deg, int n) {
  int i = blockIdx.x * blockDim.x + threadIdx.x;
  if (i < n) deg[i] = rsqrtf(deg[i]);
}

__global__ void gcn_edge_norm(const long long* __restrict__ src,
                              const long long* __restrict__ dst,
                              const float* __restrict__ ew,
                              const float* __restrict__ dinv,
                              float* __restrict__ nrm, int e) {
  int i = blockIdx.x * blockDim.x + threadIdx.x;
  if (i < e) nrm[i] = dinv[(long)src[i]] * ew[i] * dinv[(long)dst[i]];
}

__global__ void gcn_zero(float* __restrict__ p, long n) {
  long i = (long)blockIdx.x * blockDim.x + threadIdx.x;
  if (i < n) p[i] = 0.0f;
}

// ---------------------------------------------------------------------------
// Dense GEMM: C[M,N] = A[M,K] @ B[K,N] using V_WMMA_F32_16X16X4_F32.
// One wave computes one 16x16 tile; A panels staged through LDS (coalesced).
// NT = N-tiles (must divide 8 for the wave->tile mapping): 8, 4, or 1 here.
// ---------------------------------------------------------------------------
template <int K, int N>
__global__ __launch_bounds__(256) void gcn_gemm_wmma(
    const float* __restrict__ A, const float* __restrict__ B,
    float* __restrict__ C, int M) {
  constexpr int NT = (N + 15) / 16;  // 8 (N=128), 4 (N=64), 1 (N=5)
  constexpr int MP = 8 / NT;         // M-panels per 8-wave block
  __shared__ float sA[MP * 16 * K];

  const int lane = threadIdx.x & 31;
  const int w    = threadIdx.x >> 5;
  const long mBase = (long)blockIdx.x * MP * 16;

  // Cooperative stage of MP 16-row A panels (rows are contiguous in memory).
  {
    const long total = (long)MP * 16 * K;
    const long avail = ((long)M - mBase) * (long)K;
    for (long i = threadIdx.x; i < total; i += 256)
      sA[i] = (i < avail) ? A[mBase * K + i] : 0.0f;
  }
  __syncthreads();

  const int p     = w / NT;
  const int nTile = w % NT;
  const long m0   = mBase + (long)p * 16;
  if (m0 >= M) return;  // wave-uniform: EXEC stays all-ones for WMMA

  const int row = lane & 15;        // A-matrix M within tile
  const int kk  = (lane >> 4) * 2;  // lanes 0-15 -> K=0,1 ; 16-31 -> K=2,3
  const int col = lane & 15;        // B/C/D N within tile
  const float* sa = &sA[p * 16 * K];

  v8f acc = {};
  for (int k = 0; k < K; k += 4) {
    v2f a, b;
    a.x = sa[row * K + k + kk];
    a.y = sa[row * K + k + kk + 1];
    const int cn = nTile * 16 + col;
    if constexpr (N % 16 == 0) {
      b.x = B[(k + kk) * N + cn];
      b.y = B[(k + kk + 1) * N + cn];
    } else {
      const bool ok = cn < N;
      b.x = ok ? B[(k + kk) * N + cn] : 0.0f;
      b.y = ok ? B[(k + kk + 1) * N + cn] : 0.0f;
    }
    acc = __builtin_amdgcn_wmma_f32_16x16x4_f32(
        false, a, false, b, (short)0, acc, false, false);
  }

  // D layout: VGPR r, lanes 0-15 -> M=r, lanes 16-31 -> M=r+8, N=lane&15.
  const int mr = (lane >> 4) * 8;
#pragma unroll
  for (int r = 0; r < 8; ++r) {
    const int cn = nTile * 16 + col;
    if constexpr (N % 16 == 0) {
      C[(m0 + mr + r) * N + cn] = acc[r];
    } else {
      if (cn < N) C[(m0 + mr + r) * N + cn] = acc[r];
    }
  }
}

// ---------------------------------------------------------------------------
// Edge aggregation: agg[dst] += norm[e] * h[src]   (F multiple of 4)
// F/4 threads per edge, each handles 4 features (float4 gather + 4 atomics).
// ---------------------------------------------------------------------------
template <int F>
__global__ __launch_bounds__(256) void gcn_aggregate_v4(
    const long long* __restrict__ src, const long long* __restrict__ dst,
    const float* __restrict__ nrm, const float* __restrict__ h,
    float* __restrict__ agg, int E) {
  constexpr int TPE = F / 4;
  const long tid = (long)blockIdx.x * blockDim.x + threadIdx.x;
  const long e = tid / TPE;
  const int  t = (int)(tid % TPE);
  if (e >= E) return;
  const long s = (long)src[e];
  const long d = (long)dst[e];
  const float w = nrm[e];
  const float4 v = *((const float4*)(h + s * F) + t);
  float* ap = agg + d * F + t * 4;
  atomicAdd(ap + 0, w * v.x);
  atomicAdd(ap + 1, w * v.y);
  atomicAdd(ap + 2, w * v.z);
  atomicAdd(ap + 3, w * v.w);
}

template <int F>
__global__ void gcn_aggregate_small(
    const long long* __restrict__ src, const long long* __restrict__ dst,
    const float* __restrict__ nrm, const float* __restrict__ h,
    float* __restrict__ agg, int E) {
  const long e = (long)blockIdx.x * blockDim.x + threadIdx.x;
  if (e >= E) return;
  const long s = (long)src[e];
  const long d = (long)dst[e];
  const float w = nrm[e];
#pragma unroll
  for (int f = 0; f < F; ++f) atomicAdd(&agg[d * F + f], w * h[s * F + f]);
}

// ---------------------------------------------------------------------------
// out = agg + 2*dinv^2*h + bias  (optionally ReLU)
// ---------------------------------------------------------------------------
template <int F, bool RELU>
__global__ void gcn_finalize(const float* __restrict__ agg,
                             const float* __restrict__ h,
                             const float* __restrict__ dinv,
                             const float* __restrict__ bias,
                             float* __restrict__ out, int NV) {
  const long i = (long)blockIdx.x * blockDim.x + threadIdx.x;
  if (i >= (long)NV * F) return;
  const long n = i / F;
  const int  f = (int)(i % F);
  const float di = dinv[n];
  float v = agg[i] + 2.0f * di * di * h[i] + bias[f];
  if (RELU) v = fmaxf(v, 0.0f);
  out[i] = v;
}

__global__ void gcn_logsoftmax5(const float* __restrict__ logits,
                                float* __restrict__ out, int NV) {
  const int n = blockIdx.x * blockDim.x + threadIdx.x;
  if (n >= NV) return;
  float v[5], m = -3.402823466e38f;
#pragma unroll
  for (int c = 0; c < 5; ++c) { v[c] = logits[n * 5 + c]; m = fmaxf(m, v[c]); }
  float s = 0.0f;
#pragma unroll
  for (int c = 0; c < 5; ++c) s += expf(v[c] - m);
  const float lse = logf(s);
#pragma unroll
  for (int c = 0; c < 5; ++c) out[n * 5 + c] = v[c] - m - lse;
}

// ---------------------------------------------------------------------------
// Host launcher
// ---------------------------------------------------------------------------
static inline unsigned cdiv(long a, long b) { return (unsigned)((a + b - 1) / b); }

extern "C" void kernel_launch(void* const* d_in, const int* in_sizes, int n_in,
                              void* d_out, int out_size, void* d_ws, size_t ws_size,
                              hipStream_t stream) {
  const float*     x   = (const float*)d_in[0];
  const long long* ei  = (const long long*)d_in[1];
  const float*     ew  = (const float*)d_in[2];
  const float*     W1  = (const float*)d_in[3];
  const float*     b1  = (const float*)d_in[4];
  const float*     W2  = (const float*)d_in[5];
  const float*     b2  = (const float*)d_in[6];
  const float*     Wc  = (const float*)d_in[7];
  const float*     bc  = (const float*)d_in[8];

  const int E  = in_sizes[2];        // 1,600,000
  const int NV = in_sizes[0] / 128;  // 100,000
  const long long* src = ei;
  const long long* dst = ei + E;

  float* ws   = (float*)d_ws;
  float* dinv = ws;                          // NV floats
  float* nrm  = ws + 102400;                 // E floats
  float* H    = ws + 102400 + 1600000;       // NV*128 floats
  float* AGG  = H + (long)NV * 128;          // NV*128 floats
  float* ACT  = AGG + (long)NV * 128;        // NV*128 floats
  float* outp = (float*)d_out;

  // ---- degree / normalization ----
  gcn_init_deg<<<cdiv(NV, 256), 256, 0, stream>>>(dinv, NV);
  gcn_accum_deg<<<cdiv(E, 256), 256, 0, stream>>>(dst, ew, dinv, E);
  gcn_rsqrt<<<cdiv(NV, 256), 256, 0, stream>>>(dinv, NV);
  gcn_edge_norm<<<cdiv(E, 256), 256, 0, stream>>>(src, dst, ew, dinv, nrm, E);

  const int mTiles = (NV + 15) / 16;  // 6250

  // ---- layer 1: 128 -> 128, ReLU ----
  gcn_gemm_wmma<128, 128><<<cdiv(mTiles, 1), 256, 0, stream>>>(x, W1, H, NV);
  gcn_zero<<<cdiv((long)NV * 128, 256), 256, 0, stream>>>(AGG, (long)NV * 128);
  gcn_aggregate_v4<128><<<cdiv((long)E * 32, 256), 256, 0, stream>>>(src, dst, nrm, H, AGG, E);
  gcn_finalize<128, true><<<cdiv((long)NV * 128, 256), 256, 0, stream>>>(AGG, H, dinv, b1, ACT, NV);

  // ---- layer 2: 128 -> 64, ReLU ----
  gcn_gemm_wmma<128, 64><<<cdiv(mTiles, 2), 256, 0, stream>>>(ACT, W2, H, NV);
  gcn_zero<<<cdiv((long)NV * 64, 256), 256, 0, stream>>>(AGG, (long)NV * 64);
  gcn_aggregate_v4<64><<<cdiv((long)E * 16, 256), 256, 0, stream>>>(src, dst, nrm, H, AGG, E);
  gcn_finalize<64, true><<<cdiv((long)NV * 64, 256), 256, 0, stream>>>(AGG, H, dinv, b2, ACT, NV);

  // ---- classifier: 64 -> 5, log_softmax ----
  gcn_gemm_wmma<64, 5><<<cdiv(mTiles, 8), 256, 0, stream>>>(ACT, Wc, H, NV);
  gcn_zero<<<cdiv((long)NV * 5, 256), 256, 0, stream>>>(AGG, (long)NV * 5);
  gcn_aggregate_small<5><<<cdiv(E, 256), 256, 0, stream>>>(src, dst, nrm, H, AGG, E);
  gcn_finalize<5, false><<<cdiv((long)NV * 5, 256), 256, 0, stream>>>(AGG, H, dinv, bc, ACT, NV);
  gcn_logsoftmax5<<<cdiv(NV, 256), 256, 0, stream>>>(ACT, outp, NV);
}